// AdvancedTurbulenceHardCore_44341242363955
// MI455X (gfx1250) — compile-verified
//
#include <hip/hip_runtime.h>
#include <hip/hip_bf16.h>
#include <cmath>
#include <cstdint>

// ---------------------------------------------------------------------------
// SGS stress (Smagorinsky + rotational + anisotropic term) on (2,3,128^3) f32.
// Memory-bound stencil: ~151 MB traffic -> ~6.5 us roofline @ 23.3 TB/s.
// 4-wide x-vectorized, z-marching LDS ring fed by CDNA5 async global->LDS
// b128 copies (ASYNCcnt), one-plane-deep software pipelining. Block tile is
// the full W row so the x-halo is handled by in-row clamping (LDS rows stay
// 128 floats, 16B aligned -> all ds/global traffic is b128).
// ---------------------------------------------------------------------------

namespace {
constexpr int kB = 2, kC = 3, kD = 128, kH = 128, kW = 128;
constexpr int TX = 32, TY = 8, NTHR = TX * TY;   // 256 threads = 8 waves
constexpr int VX = 4;                            // voxels per thread in x
constexpr int ZCHUNK = 8, NZC = kD / ZCHUNK;     // 16 z-chunks
constexpr int ROW = kW;                          // 128 floats per LDS row
constexpr int HY = TY + 2;                       // 10 rows (y halo)
constexpr int CROW = ROW * HY;                   // 1280 floats per component
constexpr int PLANE3 = CROW * kC;                // 3840 floats per plane
constexpr int NRING = 4;                         // ring depth (pow2)
constexpr int NV4 = PLANE3 / 4;                  // 960 vec4 loads per plane
constexpr int NLDIT = (NV4 + NTHR - 1) / NTHR;   // 4 fill slots / thread
constexpr float INV2DX = 64.0f;                  // 1/(2*dx), dx = 1/128
}  // namespace

#if defined(__gfx1250__) && __has_builtin(__builtin_amdgcn_global_load_async_to_lds_b128)
#define SGS_ASYNC 1
#else
#define SGS_ASYNC 0
#if defined(__HIP_DEVICE_COMPILE__)
#warning "async global->LDS b128 builtin unavailable; sync fallback in use"
#endif
#endif

typedef int v4i_t __attribute__((vector_size(4 * sizeof(int))));
typedef __attribute__((address_space(1))) v4i_t g_v4i_t;
typedef __attribute__((address_space(3))) v4i_t l_v4i_t;

__device__ __forceinline__ int iclampi(int v, int hi) {
  v = v < 0 ? 0 : v;
  return v > hi ? hi : v;
}

__device__ __forceinline__ float softplus_f(float x) {
  return (x > 20.0f) ? x : log1pf(expf(x));
}

__device__ __forceinline__ void sgs_wait_async() {
#if SGS_ASYNC
#if __has_builtin(__builtin_amdgcn_s_wait_asynccnt)
  __builtin_amdgcn_s_wait_asynccnt(0);
#else
  asm volatile("s_wait_asynccnt 0" ::: "memory");
#endif
#endif
}

__global__ __launch_bounds__(NTHR) void sgs_stress_kernel(
    const float* __restrict__ vel, const float* __restrict__ cs_p,
    const float* __restrict__ cr_p, const float* __restrict__ ca_p,
    float* __restrict__ out) {
  __shared__ float smem[NRING * PLANE3];  // 4 * 3840 * 4B = 60 KB

  const int tx = threadIdx.x, ty = threadIdx.y;
  const int tid = ty * TX + tx;
  const int y0 = blockIdx.y * TY;
  const int b  = blockIdx.z / NZC;
  const int z0 = (blockIdx.z % NZC) * ZCHUNK;

  const long HW = (long)kH * kW;
  const long N  = (long)kD * HW;
  const float* vb = vel + (long)b * kC * N;

  // Uniform coefficients (scalarized by the compiler; once per block).
  const float cs = softplus_f(*cs_p) * 0.2f;
  const float cr = softplus_f(*cr_p) * 0.1f;
  const float ca = softplus_f(*ca_p) * 0.05f;
  const float cs2 = cs * cs;  // * delta^2 (= 1)
  const float cr2 = cr * cr;

  // z-invariant plane-fill mapping: vec4 index q -> (comp, yy, 4*xq).
  long goff[NLDIT];
  int  lidx[NLDIT];
  bool vld[NLDIT];
#pragma unroll
  for (int it = 0; it < NLDIT; ++it) {
    const int q = tid + it * NTHR;
    vld[it] = (q < NV4);
    const int qq = vld[it] ? q : 0;
    const int comp = qq / (NV4 / kC);        // / 320
    const int r    = qq - comp * (NV4 / kC);
    const int yy   = r / (ROW / 4);          // / 32
    const int xq   = r - yy * (ROW / 4);
    const int gy = iclampi(y0 - 1 + yy, kH - 1);  // edge-clamp == pad 'edge'
    goff[it] = (long)comp * N + (long)gy * kW + 4 * xq;
    lidx[it] = comp * CROW + yy * ROW + 4 * xq;
  }

  auto slot_of = [](int z) { return (z + NRING) & (NRING - 1); };

  auto issue_plane = [&](int zq, int slot) {
    const long zoff = (long)iclampi(zq, kD - 1) * HW;
    float* dstbase = &smem[slot * PLANE3];
#pragma unroll
    for (int it = 0; it < NLDIT; ++it) {
      if (!vld[it]) continue;
      const float* src = vb + goff[it] + zoff;
      float* dst = dstbase + lidx[it];
#if SGS_ASYNC
      __builtin_amdgcn_global_load_async_to_lds_b128(
          (g_v4i_t*)src, (l_v4i_t*)dst, 0, 0);
#else
      *(float4*)dst = *(const float4*)src;
#endif
    }
  };

  // Prologue: planes z0-1, z0, z0+1.
  issue_plane(z0 - 1, slot_of(z0 - 1));
  issue_plane(z0,     slot_of(z0));
  issue_plane(z0 + 1, slot_of(z0 + 1));
  sgs_wait_async();
  __syncthreads();

  const int yy = ty + 1;            // LDS row of this thread's y
  const int x4 = VX * tx;           // first x of this thread
  const int xm = (x4 == 0) ? 0 : (x4 - 1);             // left edge clamp
  const int xp = (x4 + VX > kW - 1) ? (kW - 1) : (x4 + VX);  // right edge
  const long obase0 = (long)b * 6 * N + (long)(y0 + ty) * kW + x4;

  for (int z = z0; z < z0 + ZCHUNK; ++z) {
    if (z + 1 < z0 + ZCHUNK) issue_plane(z + 2, slot_of(z + 2));

    const float* pm = &smem[slot_of(z - 1) * PLANE3];
    const float* pc = &smem[slot_of(z)     * PLANE3];
    const float* pp = &smem[slot_of(z + 1) * PLANE3];

    // g*[c][j] = d v_c / d {x,y,z} at voxel x4+j
    float gx[kC][VX], gy[kC][VX], gz[kC][VX];
#pragma unroll
    for (int c = 0; c < kC; ++c) {
      const float* rowc = pc + c * CROW + yy * ROW;
      const float4 c4 = *(const float4*)(rowc + x4);
      const float lm = rowc[xm];
      const float rp = rowc[xp];
      const float4 u4 = *(const float4*)(pc + c * CROW + (yy - 1) * ROW + x4);
      const float4 d4 = *(const float4*)(pc + c * CROW + (yy + 1) * ROW + x4);
      const float4 m4 = *(const float4*)(pm + c * CROW + yy * ROW + x4);
      const float4 p4 = *(const float4*)(pp + c * CROW + yy * ROW + x4);

      gx[c][0] = (c4.y - lm)   * INV2DX;
      gx[c][1] = (c4.z - c4.x) * INV2DX;
      gx[c][2] = (c4.w - c4.y) * INV2DX;
      gx[c][3] = (rp   - c4.z) * INV2DX;

      const float ua[VX] = {u4.x, u4.y, u4.z, u4.w};
      const float da[VX] = {d4.x, d4.y, d4.z, d4.w};
      const float ma[VX] = {m4.x, m4.y, m4.z, m4.w};
      const float pa[VX] = {p4.x, p4.y, p4.z, p4.w};
#pragma unroll
      for (int j = 0; j < VX; ++j) {
        gy[c][j] = (da[j] - ua[j]) * INV2DX;
        gz[c][j] = (pa[j] - ma[j]) * INV2DX;
      }
    }

    float t[6][VX];
#pragma unroll
    for (int j = 0; j < VX; ++j) {
      const float S00 = gx[0][j], S11 = gy[1][j], S22 = gz[2][j];
      const float S01 = 0.5f * (gy[0][j] + gx[1][j]);
      const float S02 = 0.5f * (gz[0][j] + gx[2][j]);
      const float S12 = 0.5f * (gz[1][j] + gy[2][j]);
      const float O01 = 0.5f * (gy[0][j] - gx[1][j]);
      const float O02 = 0.5f * (gz[0][j] - gx[2][j]);
      const float O12 = 0.5f * (gz[1][j] - gy[2][j]);

      const float Ssq = S00 * S00 + S11 * S11 + S22 * S22 +
                        2.0f * (S01 * S01 + S02 * S02 + S12 * S12);
      const float Osq = 2.0f * (O01 * O01 + O02 * O02 + O12 * O12);
      const float m2nud = -2.0f * (cs2 * sqrtf(2.0f * Ssq + 1e-8f));
      const float m2nur = -2.0f * (cr2 * sqrtf(2.0f * Osq + 1e-8f));

      const float SS00 = S00 * S00 + S01 * S01 + S02 * S02;
      const float SS11 = S01 * S01 + S11 * S11 + S12 * S12;
      const float SS22 = S02 * S02 + S12 * S12 + S22 * S22;
      const float SS01 = S00 * S01 + S01 * S11 + S02 * S12;
      const float SS02 = S00 * S02 + S01 * S12 + S02 * S22;
      const float SS12 = S01 * S02 + S11 * S12 + S12 * S22;
      const float tr3  = (1.0f / 3.0f) * Ssq;

      t[0][j] = m2nud * S00 + ca * (SS00 - tr3);
      t[1][j] = m2nud * S11 + ca * (SS11 - tr3);
      t[2][j] = m2nud * S22 + ca * (SS22 - tr3);
      t[3][j] = m2nud * S01 + m2nur * O01 + ca * SS01;
      t[4][j] = m2nud * S02 + m2nur * O02 + ca * SS02;
      t[5][j] = m2nud * S12 + m2nur * O12 + ca * SS12;
    }

    const long ob = obase0 + (long)z * HW;
#pragma unroll
    for (int k = 0; k < 6; ++k) {
      *(float4*)(out + ob + (long)k * N) =
          make_float4(t[k][0], t[k][1], t[k][2], t[k][3]);
    }

    sgs_wait_async();
    __syncthreads();
  }
}

extern "C" void kernel_launch(void* const* d_in, const int* in_sizes, int n_in,
                              void* d_out, int out_size, void* d_ws,
                              size_t ws_size, hipStream_t stream) {
  (void)in_sizes; (void)n_in; (void)out_size; (void)d_ws; (void)ws_size;
  dim3 block(TX, TY, 1);
  dim3 grid(1, kH / TY, kB * NZC);  // (1, 16, 32) = 512 blocks, 8 waves each
  sgs_stress_kernel<<<grid, block, 0, stream>>>(
      (const float*)d_in[0], (const float*)d_in[1], (const float*)d_in[2],
      (const float*)d_in[3], (float*)d_out);
}